// TFVectorRepRandomizerPool_88923002896591
// MI455X (gfx1250) — compile-verified
//
#include <hip/hip_runtime.h>
#include <stdint.h>

// TFVectorRepRandomizerPool gather for MI455X (gfx1250).
//
// Pure data movement: ~16 MB gathered reads + 16 MB streamed writes
// => ~1.4 us at 23.3 TB/s. No FLOPs -> no WMMA; the CDNA5 feature that
// matters is the async global<->LDS data mover (ASYNCcnt) + NT cache hints
// (dataset is 51 GB, rows touched once -> don't pollute L2).
//
// One wave32 per 1KB output row: lane l moves bytes [l*16, l*16+16) and
// [512 + l*16, ...) via two b128 async ops (inst offset advances BOTH the
// LDS and global address per the ISA pseudocode).

#define POOL_SIZE 16
#define DIM 256
#define WAVES_PER_BLOCK 8
#define BLOCK_THREADS 256

__device__ __forceinline__ uint32_t rotl32(uint32_t x, int n) {
  return (x << n) | (x >> (32 - n));
}

// Exact replica of JAX threefry2x32 with key = jax.random.key(42) -> (0, 42).
__device__ __forceinline__ void threefry2x32_key42(uint32_t x0, uint32_t x1,
                                                   uint32_t& o0, uint32_t& o1) {
  const uint32_t ks0 = 0u;
  const uint32_t ks1 = 42u;
  const uint32_t ks2 = 0x1BD11BDAu ^ ks0 ^ ks1;
  uint32_t v0 = x0 + ks0;
  uint32_t v1 = x1 + ks1;
#define TF_R4(a, b, c, d)                                                      \
  v0 += v1; v1 = rotl32(v1, a); v1 ^= v0;                                      \
  v0 += v1; v1 = rotl32(v1, b); v1 ^= v0;                                      \
  v0 += v1; v1 = rotl32(v1, c); v1 ^= v0;                                      \
  v0 += v1; v1 = rotl32(v1, d); v1 ^= v0;
  TF_R4(13, 15, 26, 6)  v0 += ks1; v1 += ks2 + 1u;
  TF_R4(17, 29, 16, 24) v0 += ks2; v1 += ks0 + 2u;
  TF_R4(13, 15, 26, 6)  v0 += ks0; v1 += ks1 + 3u;
  TF_R4(17, 29, 16, 24) v0 += ks1; v1 += ks2 + 4u;
  TF_R4(13, 15, 26, 6)  v0 += ks2; v1 += ks0 + 5u;
#undef TF_R4
  o0 = v0;
  o1 = v1;
}

__global__ __launch_bounds__(BLOCK_THREADS)
void pool_gather_kernel(const float* __restrict__ vectors,
                        const int* __restrict__ lengths,
                        const int* __restrict__ objs,
                        float* __restrict__ out,
                        int nrows, int halfn) {
  __shared__ float lbuf[WAVES_PER_BLOCK * DIM];  // 8 KB: 1 KB per wave

  const int tid  = (int)threadIdx.x;
  const int wave = tid >> 5;
  const int lane = tid & 31;
  const int row  = (int)blockIdx.x * WAVES_PER_BLOCK + wave;
  if (row >= nrows) return;  // wave-uniform exit

  // JAX random_bits layout: counts = iota(n) split into two halves; the pair
  // (i, i + n/2) is encrypted once; out[i] = enc.x0, out[i + n/2] = enc.x1.
  const uint32_t base = (uint32_t)(row < halfn ? row : row - halfn);
  uint32_t o0, o1;
  threefry2x32_key42(base, base + (uint32_t)halfn, o0, o1);
  const uint32_t bits = (row < halfn) ? o0 : o1;
  // uniform [0,1): bitcast(bits>>9 | 0x3f800000) - 1.0
  const float u = __uint_as_float((bits >> 9) | 0x3F800000u) - 1.0f;

  const int pool = objs[row];
  const int len  = lengths[pool];
  int idx = min((int)(u * (float)len), len - 1);  // trunc-toward-zero == astype(int32)
  if (len == 0) idx = 0;

  const float* src = vectors + ((size_t)pool * POOL_SIZE + (size_t)idx) * DIM;
  float* dst = out + (size_t)row * DIM;

  // LDS byte offset of this wave's staging buffer (generic addr low 32 bits
  // == LDS offset per the flat->LDS aperture mapping).
  const uint32_t lds0 =
      (uint32_t)(uintptr_t)(&lbuf[wave * DIM]) + (uint32_t)(lane * 16);
  const uint64_t s0 = (uint64_t)(uintptr_t)src + (uint64_t)(lane * 16);
  const uint64_t d0 = (uint64_t)(uintptr_t)dst + (uint64_t)(lane * 16);

  // HBM -> LDS (async DMA path, non-temporal: rows touched ~once)
  asm volatile("global_load_async_to_lds_b128 %0, %1, off th:TH_LOAD_NT"
               :: "v"(lds0), "v"(s0) : "memory");
  asm volatile("global_load_async_to_lds_b128 %0, %1, off offset:512 th:TH_LOAD_NT"
               :: "v"(lds0), "v"(s0) : "memory");
  asm volatile("s_wait_asynccnt 0" ::: "memory");
  // LDS -> HBM (streaming store, never re-read by this kernel)
  asm volatile("global_store_async_from_lds_b128 %0, %1, off th:TH_STORE_NT"
               :: "v"(d0), "v"(lds0) : "memory");
  asm volatile("global_store_async_from_lds_b128 %0, %1, off offset:512 th:TH_STORE_NT"
               :: "v"(d0), "v"(lds0) : "memory");
  asm volatile("s_wait_asynccnt 0" ::: "memory");
}

extern "C" void kernel_launch(void* const* d_in, const int* in_sizes, int n_in,
                              void* d_out, int out_size, void* d_ws, size_t ws_size,
                              hipStream_t stream) {
  const float* vectors = (const float*)d_in[0];  // [N_POOLS, 16, 256] f32
  const int*   lengths = (const int*)d_in[1];    // [N_POOLS] i32
  const int*   objs    = (const int*)d_in[2];    // [BATCH] i32
  float*       out     = (float*)d_out;          // [BATCH, 256] f32

  const int nrows = in_sizes[2];       // BATCH (16384)
  const int halfn = nrows >> 1;        // JAX counter split point
  const int grid  = (nrows + WAVES_PER_BLOCK - 1) / WAVES_PER_BLOCK;

  pool_gather_kernel<<<grid, BLOCK_THREADS, 0, stream>>>(
      vectors, lengths, objs, out, nrows, halfn);
}